// TemporalSimilarityLoss_4320737099939
// MI455X (gfx1250) — compile-verified
//
#include <hip/hip_runtime.h>
#include <math.h>

#define N_TRAJ 64
#define T_STEP 64
#define D_FEAT 256
#define NT     (N_TRAJ * T_STEP)   // 4096
#define M_ROWS (2 * NT)            // 8192
#define EPS_F  1e-8f

// pos pairs: 64 * 128 * 127 ; neg pairs: 8192^2 - 64*128*128
#define POS_CNT 1040384.0f
#define NEG_CNT 66060288.0f

typedef __attribute__((ext_vector_type(2))) float v2f;
typedef __attribute__((ext_vector_type(8))) float v8f;

// ---------------------------------------------------------------------------
// Kernel 1: per-row L2 norm, one wave32 per row (8 waves / 256-thread block).
// ---------------------------------------------------------------------------
__global__ void row_norms_kernel(const float* __restrict__ z,
                                 float* __restrict__ norm) {
  const int wave = threadIdx.x >> 5;
  const int lane = threadIdx.x & 31;
  const int row  = blockIdx.x * (blockDim.x >> 5) + wave;
  if (row >= M_ROWS) return;

  const float* zr = z + (size_t)row * D_FEAT;
  float s = 0.0f;
#pragma unroll
  for (int e = 0; e < D_FEAT / 32; ++e) {
    float v = zr[lane + 32 * e];
    s += v * v;
  }
#pragma unroll
  for (int off = 16; off > 0; off >>= 1)
    s += __shfl_xor(s, off, 32);

  if (lane == 0) norm[row] = sqrtf(s);
}

// ---------------------------------------------------------------------------
// Kernel 2: trajectory sums of normalized rows. Block t handles trajectory t
// (rows t*64..t*64+63 and 4096+t*64..+63); thread d handles feature d.
// ---------------------------------------------------------------------------
__global__ void traj_sums_kernel(const float* __restrict__ z,
                                 const float* __restrict__ norm,
                                 float* __restrict__ sums) {
  const int t = blockIdx.x;     // 0..63
  const int d = threadIdx.x;    // 0..255
  float acc = 0.0f;
#pragma unroll 4
  for (int j = 0; j < 2 * T_STEP; ++j) {
    const int row = (j < T_STEP) ? (t * T_STEP + j)
                                 : (NT + t * T_STEP + (j - T_STEP));
    const float rn = 1.0f / fmaxf(norm[row], EPS_F);
    acc += z[(size_t)row * D_FEAT + d] * rn;
  }
  sums[(size_t)t * D_FEAT + d] = acc;
}

// ---------------------------------------------------------------------------
// Kernel 3: 64x64 Gram of trajectory sums via V_WMMA_F32_16X16X4_F32.
// 1 block x 512 threads = 16 waves; wave w computes tile (ti=w>>2, tj=w&3),
// looping K=256 in steps of 4. Then reduce total-sum / trace / diag and emit
// the two scalars.
//
// ISA 7.12.2 fragment layout (32-bit, 16x4 A):
//   lane l: m = l&15 ; VGPR0 = K = 2*(l>>4), VGPR1 = K = 2*(l>>4)+1
// C/D 16x16: lane l: n = l&15 ; VGPR v: m = v + 8*(l>>4)
// B (4x16) mirrors A with N on lanes; since B = Sums^T the per-lane loads
// are identical to A's with the tile-row offset swapped.
// ---------------------------------------------------------------------------
__global__ void gram_finalize_kernel(const float* __restrict__ sums,
                                     const float* __restrict__ norm,
                                     float* __restrict__ out) {
  __shared__ float sTot[512];
  __shared__ float sTr[512];
  __shared__ float sDg[512];

  const int tid  = threadIdx.x;
  const int wave = tid >> 5;
  const int lane = tid & 31;
  const int ti   = wave >> 2;      // 0..3
  const int tj   = wave & 3;       // 0..3
  const int half = lane >> 4;      // 0 or 1
  const int l15  = lane & 15;
  const int kb   = half * 2;

  const float* arow = sums + (size_t)(ti * 16 + l15) * D_FEAT;
  const float* brow = sums + (size_t)(tj * 16 + l15) * D_FEAT;

  v8f c = {0.f, 0.f, 0.f, 0.f, 0.f, 0.f, 0.f, 0.f};
#pragma unroll 4
  for (int k = 0; k < D_FEAT; k += 4) {
    v2f a = *(const v2f*)(arow + k + kb);
    v2f b = *(const v2f*)(brow + k + kb);
    // D = A * B + C, fp32 16x16x4
    c = __builtin_amdgcn_wmma_f32_16x16x4_f32(
        /*neg_a=*/false, a, /*neg_b=*/false, b,
        /*c_mod=*/(short)0, c, /*reuse_a=*/false, /*reuse_b=*/false);
  }

  // Per-lane partials: total sum of Gram, and trace (global row == col).
  float tot = 0.0f, tr = 0.0f;
#pragma unroll
  for (int v = 0; v < 8; ++v) {
    tot += c[v];
    if (ti == tj && (v + 8 * half) == l15) tr += c[v];
  }

  // Diagonal of logits: (||z_i|| / max(||z_i||, eps))^2
  float dg = 0.0f;
  for (int i = tid; i < M_ROWS; i += 512) {
    float n = norm[i];
    float r = n / fmaxf(n, EPS_F);
    dg += r * r;
  }

  sTot[tid] = tot; sTr[tid] = tr; sDg[tid] = dg;
  __syncthreads();
  for (int s = 256; s > 0; s >>= 1) {
    if (tid < s) {
      sTot[tid] += sTot[tid + s];
      sTr[tid]  += sTr[tid + s];
      sDg[tid]  += sDg[tid + s];
    }
    __syncthreads();
  }

  if (tid == 0) {
    const float total = sTot[0];   // sum over ALL pairs of cosine sim
    const float trace = sTr[0];    // sum over same-trajectory pairs (incl diag)
    const float diag  = sDg[0];    // sum over diagonal
    out[0] = (trace - diag) / POS_CNT;   // positive_sim
    out[1] = (total - trace) / NEG_CNT;  // negative_sim
  }
}

// ---------------------------------------------------------------------------
extern "C" void kernel_launch(void* const* d_in, const int* in_sizes, int n_in,
                              void* d_out, int out_size, void* d_ws, size_t ws_size,
                              hipStream_t stream) {
  const float* z = (const float*)d_in[0];   // [8192, 256] f32
  // d_in[1] = done [64,64] int32 : provably unused (positive_mask == 1 - tm + eye)
  float* out = (float*)d_out;               // [2] f32: (positive_sim, negative_sim)

  float* ws    = (float*)d_ws;
  float* norm  = ws;                        // 8192 floats
  float* sums  = ws + M_ROWS;               // 64*256 floats

  // 1) per-row norms: 8192 rows, 8 waves/block
  row_norms_kernel<<<M_ROWS / 8, 256, 0, stream>>>(z, norm);
  // 2) trajectory sums of normalized rows
  traj_sums_kernel<<<N_TRAJ, D_FEAT, 0, stream>>>(z, norm, sums);
  // 3) fp32 WMMA Gram + finalize
  gram_finalize_kernel<<<1, 512, 0, stream>>>(sums, norm, out);
}